// Encoder_51041391346271
// MI455X (gfx1250) — compile-verified
//
#include <hip/hip_runtime.h>

// GRU cell step, MI455X (gfx1250, wave32).
// Bandwidth-bound: 100.7 MB of f32 weights -> ~4.3 us floor @ 23.3 TB/s.
// Mat-vec done via V_WMMA_F32_16X16X4_F32 (B = vector chunk replicated
// across N columns; compute waste is free under the roofline).

typedef __attribute__((ext_vector_type(2))) float v2f;
typedef __attribute__((ext_vector_type(8))) float v8f;

#define Hdim 2048
#define ROWS_PER_BLOCK 16
#define WAVES 8
#define K_PER_WAVE (Hdim / WAVES) // 256
#define IH_BLOCKS ((3 * Hdim) / ROWS_PER_BLOCK) // 384

__global__ __launch_bounds__(256) void gru_matvec_wmma(
    const int* __restrict__ inp,
    const float* __restrict__ hidden,   // [H]
    const float* __restrict__ emb,      // [VOCAB, H]
    const float* __restrict__ W_ih,     // [3H, H] row-major
    const float* __restrict__ W_hh,     // [3H, H] row-major
    float* __restrict__ gout)           // [6H]: gi in [0,3H), gh in [3H,6H)
{
    __shared__ float red[WAVES * ROWS_PER_BLOCK];

    const int tid  = threadIdx.x;
    const int lane = tid & 31;
    const int wave = tid >> 5;
    const int half = lane >> 4;   // lane-half selects K pair {0,1} vs {2,3}
    const int m    = lane & 15;   // A-matrix row within the 16-row tile

    const int  blk      = blockIdx.x;
    const bool is_ih    = blk < IH_BLOCKS;
    const int  row_base = (is_ih ? blk : blk - IH_BLOCKS) * ROWS_PER_BLOCK;

    const float* W   = is_ih ? W_ih : W_hh;
    const float* vec = is_ih ? (emb + (size_t)inp[0] * Hdim) : hidden;

    // Each lane streams one weight row (lanes l and l+16 share row l&15,
    // consuming complementary K pairs -> full cacheline consumption).
    const float* Arow = W + (size_t)(row_base + m) * Hdim;

    const int k0 = wave * K_PER_WAVE;

    // Two accumulators to break the WMMA D->C dependency chain.
    v8f acc0 = {};
    v8f acc1 = {};

    for (int k = k0; k < k0 + K_PER_WAVE; k += 8) {
        const int kka = k + 2 * half;       // K pair for step 0
        const int kkb = k + 4 + 2 * half;   // K pair for step 1

        // A layout (16x4 f32): VGPR0 = K0 (lanes0-15)/K2 (lanes16-31),
        //                      VGPR1 = K1/K3  -> one b64 load per lane.
        v2f a0 = *(const v2f*)(Arow + kka);
        v2f a1 = *(const v2f*)(Arow + kkb);
        // B (4x16): lane%16 = column N; replicate the vector into all columns,
        // same per-lane K addressing as A.
        v2f b0 = *(const v2f*)(vec + kka);
        v2f b1 = *(const v2f*)(vec + kkb);

        acc0 = __builtin_amdgcn_wmma_f32_16x16x4_f32(
            false, a0, false, b0, (short)0, acc0, false, false);
        acc1 = __builtin_amdgcn_wmma_f32_16x16x4_f32(
            false, a1, false, b1, (short)0, acc1, false, false);
    }

    const v8f acc = acc0 + acc1;

    // Every column of D holds the same partial dot products. Column 0 lives in
    // lane 0 (rows 0-7 in c.s0..s7) and lane 16 (rows 8-15).
    if (m == 0) {
        float* dst = &red[wave * ROWS_PER_BLOCK + half * 8];
        dst[0] = acc.s0; dst[1] = acc.s1; dst[2] = acc.s2; dst[3] = acc.s3;
        dst[4] = acc.s4; dst[5] = acc.s5; dst[6] = acc.s6; dst[7] = acc.s7;
    }
    __syncthreads();

    // Reduce the 8 wave K-partials for each of the 16 rows.
    if (tid < ROWS_PER_BLOCK) {
        float s = 0.f;
        #pragma unroll
        for (int w = 0; w < WAVES; ++w) s += red[w * ROWS_PER_BLOCK + tid];
        const size_t out_idx = (is_ih ? 0 : (size_t)(3 * Hdim)) + row_base + tid;
        gout[out_idx] = s;
    }
}

__global__ __launch_bounds__(256) void gru_gates(
    const float* __restrict__ g,       // ws: gi [0,3H), gh [3H,6H)
    const float* __restrict__ hidden,  // [H]
    const float* __restrict__ b_ih,    // [3H]
    const float* __restrict__ b_hh,    // [3H]
    float* __restrict__ out)           // [2H]: tuple (out, out) concatenated
{
    const int i = blockIdx.x * blockDim.x + threadIdx.x;
    if (i >= Hdim) return;

    const float i_r = g[i]             + b_ih[i];
    const float i_z = g[Hdim + i]      + b_ih[Hdim + i];
    const float i_n = g[2 * Hdim + i]  + b_ih[2 * Hdim + i];
    const float h_r = g[3 * Hdim + i]             + b_hh[i];
    const float h_z = g[3 * Hdim + Hdim + i]      + b_hh[Hdim + i];
    const float h_n = g[3 * Hdim + 2 * Hdim + i]  + b_hh[2 * Hdim + i];

    const float r = 1.0f / (1.0f + expf(-(i_r + h_r)));
    const float z = 1.0f / (1.0f + expf(-(i_z + h_z)));
    const float n = tanhf(i_n + r * h_n);
    const float h_new = (1.0f - z) * n + z * hidden[i];

    out[i] = h_new;
    out[Hdim + i] = h_new;
}

extern "C" void kernel_launch(void* const* d_in, const int* in_sizes, int n_in,
                              void* d_out, int out_size, void* d_ws, size_t ws_size,
                              hipStream_t stream) {
    const int*   inp    = (const int*)  d_in[0];
    const float* hidden = (const float*)d_in[1];
    const float* emb    = (const float*)d_in[2];
    const float* W_ih   = (const float*)d_in[3];
    const float* W_hh   = (const float*)d_in[4];
    const float* b_ih   = (const float*)d_in[5];
    const float* b_hh   = (const float*)d_in[6];

    float* out = (float*)d_out;
    float* g   = (float*)d_ws;   // needs 6H floats = 48 KB scratch

    gru_matvec_wmma<<<dim3(2 * IH_BLOCKS), dim3(256), 0, stream>>>(
        inp, hidden, emb, W_ih, W_hh, g);
    gru_gates<<<dim3((Hdim + 255) / 256), dim3(256), 0, stream>>>(
        g, hidden, b_ih, b_hh, out);
}